// WHT1D_42322607734920
// MI455X (gfx1250) — compile-verified
//
#include <hip/hip_runtime.h>
#include <hip/hip_bf16.h>

// ---------------------------------------------------------------------------
// WHT1D pipeline for MI455X (gfx1250), all matmuls on V_WMMA_F32_16X16X4_F32.
// H4096 = H64 (x) H64 (Sylvester Kronecker factorization), so each length-
// 4096 WHT is H64 * X(64x64) * H64.  H64 is materialized once per workgroup
// in LDS (padded stride 65) so fragment builds are ds_load_2addr pairs, not
// popcount/cndmask VALU chains.  Each wave owns two tiles sharing the A
// fragment -> two independent WMMA accumulation chains.
// ---------------------------------------------------------------------------

typedef __attribute__((ext_vector_type(2))) float v2f;
typedef __attribute__((ext_vector_type(8))) float v8f;

#define NB 8      // batches
#define NC 64     // channels
#define NPAD 4096
#define NLEN 4000
#define NP 4      // p dimension
#define HS 65     // padded LDS row stride

__device__ __forceinline__ v8f wmma4f32(v2f a, v2f b, v8f c) {
    // (neg_a, A, neg_b, B, c_mod, C, reuse_a, reuse_b)
    return __builtin_amdgcn_wmma_f32_16x16x4_f32(
        false, a, false, b, (short)0, c, false, false);
}

// One-time H64 materialization: H[i][j] = (-1)^popcount(i&j), stride-65 LDS.
__device__ __forceinline__ void fill_H(float* Hs, int tid) {
    for (int idx = tid; idx < 4096; idx += 256) {
        const int i = idx >> 6, j = idx & 63;
        Hs[i * HS + j] = __int_as_float(0x3f800000u | ((__popc(i & j) & 1u) << 31));
    }
}

// ---------------------------------------------------------------------------
// Kernel 1: forward WHT.  One workgroup per (b,i) row: pad to 4096, reshape
// 64x64, Y = H * X * H, write f1[b][i][:] (n = 64*n1 + n2).
// ---------------------------------------------------------------------------
__global__ void __launch_bounds__(256) wht_fwd(const float* __restrict__ x,
                                               float* __restrict__ f1) {
    __shared__ float Xs[64 * 64];     // stride 64 (B-frags read along rows)
    __shared__ float Ts[64 * HS];     // stride 65 (A-frags read down columns)
    __shared__ float Hs[64 * HS];     // H64, stride 65 (serves both patterns)

    const int row = blockIdx.x;       // b*64 + i
    const int tid = threadIdx.x;
    const float* xr = x + (size_t)row * NLEN;

    fill_H(Hs, tid);
    for (int idx = tid; idx < NPAD; idx += 256)
        Xs[idx] = (idx < NLEN) ? xr[idx] : 0.0f;
    __syncthreads();

    const int wave  = tid >> 5;
    const int lane  = tid & 31;
    const int laneN = lane & 15;
    const int kHalf = (lane >> 4) << 1;          // 0 or 2
    const int mt    = wave & 3;                  // tile row
    const int nt0   = (wave >> 2) * 2;           // tile col pair {nt0, nt0+1}

    // ---- stage 1: T = H @ X  (A = H from LDS, shared across the pair) ----
    {
        v8f c0 = {}, c1 = {};
        const int mrow = mt * 16 + laneN;
#pragma unroll
        for (int kt = 0; kt < 16; ++kt) {
            const int k0 = kt * 4 + kHalf;
            v2f a;                                 // H[m][k0], H[m][k0+1]
            a.x = Hs[mrow * HS + k0];
            a.y = Hs[mrow * HS + k0 + 1];
            v2f b0, b1;                            // X[k][n] rows
            b0.x = Xs[k0 * 64 + nt0 * 16 + laneN];
            b0.y = Xs[(k0 + 1) * 64 + nt0 * 16 + laneN];
            b1.x = Xs[k0 * 64 + (nt0 + 1) * 16 + laneN];
            b1.y = Xs[(k0 + 1) * 64 + (nt0 + 1) * 16 + laneN];
            c0 = wmma4f32(a, b0, c0);
            c1 = wmma4f32(a, b1, c1);
        }
        const int rbase = mt * 16 + (lane >> 4) * 8;
        const int ci0   = nt0 * 16 + laneN;
#pragma unroll
        for (int r = 0; r < 8; ++r) {
            Ts[(rbase + r) * HS + ci0]      = c0[r];
            Ts[(rbase + r) * HS + ci0 + 16] = c1[r];
        }
    }
    __syncthreads();

    // ---- stage 2: Y = T @ H  (A = T from LDS, B = H from LDS) ----
    {
        float* f1r = f1 + (size_t)row * NPAD;
        v8f c0 = {}, c1 = {};
        const int mrow = mt * 16 + laneN;
#pragma unroll
        for (int kt = 0; kt < 16; ++kt) {
            const int k0 = kt * 4 + kHalf;
            v2f a;                                 // T[m][k0], T[m][k0+1]
            a.x = Ts[mrow * HS + k0];
            a.y = Ts[mrow * HS + k0 + 1];
            v2f b0, b1;                            // H[k][n] rows
            b0.x = Hs[k0 * HS + nt0 * 16 + laneN];
            b0.y = Hs[(k0 + 1) * HS + nt0 * 16 + laneN];
            b1.x = Hs[k0 * HS + (nt0 + 1) * 16 + laneN];
            b1.y = Hs[(k0 + 1) * HS + (nt0 + 1) * 16 + laneN];
            c0 = wmma4f32(a, b0, c0);
            c1 = wmma4f32(a, b1, c1);
        }
        const int rbase = mt * 16 + (lane >> 4) * 8;   // n1
        const int ci0   = nt0 * 16 + laneN;            // n2
#pragma unroll
        for (int r = 0; r < 8; ++r) {
            f1r[(size_t)(rbase + r) * 64 + ci0]      = c0[r];
            f1r[(size_t)(rbase + r) * 64 + ci0 + 16] = c1[r];
        }
    }
}

// ---------------------------------------------------------------------------
// Kernel 2: channel mixing.  One workgroup per (b, 64-wide n block).
// For each p: C(64x64) = W[p] @ (v[p,n] * f1[b,:,n]), soft-threshold by
// |T[p,n]|, accumulate over p -> f6[b][o][n].  W[p] staged in LDS per p.
// ---------------------------------------------------------------------------
__global__ void __launch_bounds__(256) chmix(const float* __restrict__ f1,
                                             const float* __restrict__ W,
                                             const float* __restrict__ Tthr,
                                             const float* __restrict__ v,
                                             float* __restrict__ f6) {
    __shared__ float Zs[64 * 64];     // f1 block, stride 64
    __shared__ float Ws[64 * HS];     // W[p], stride 65 (column-wise A reads)

    const int b   = blockIdx.x >> 6;    // 0..7
    const int nb  = blockIdx.x & 63;    // 0..63 (64-col block)
    const int tid = threadIdx.x;

    const float* f1b = f1 + (size_t)b * NC * NPAD + (size_t)nb * 64;
    for (int idx = tid; idx < 4096; idx += 256) {
        const int i = idx >> 6, nn = idx & 63;
        Zs[idx] = f1b[(size_t)i * NPAD + nn];
    }

    const int wave  = tid >> 5;
    const int lane  = tid & 31;
    const int laneN = lane & 15;
    const int kHalf = (lane >> 4) << 1;
    const int ot    = wave & 3;
    const int nt0   = (wave >> 2) * 2;

    const int ncol0 = nb * 64 + nt0 * 16 + laneN;
    const int ncol1 = ncol0 + 16;
    const int orow  = ot * 16 + laneN;

    v8f acc0 = {}, acc1 = {};
    for (int p = 0; p < NP; ++p) {
        // stage W[p] into LDS (stride 65)
        __syncthreads();                       // protect Ws/Zs from prev use
        const float* Wp = W + p * NC * NC;
        for (int idx = tid; idx < 4096; idx += 256) {
            const int o = idx >> 6, k = idx & 63;
            Ws[o * HS + k] = Wp[idx];
        }
        __syncthreads();

        const float vp0 = v[p * NPAD + ncol0];
        const float vp1 = v[p * NPAD + ncol1];
        const float tp0 = fabsf(Tthr[p * NPAD + ncol0]);
        const float tp1 = fabsf(Tthr[p * NPAD + ncol1]);

        v8f c0 = {}, c1 = {};
#pragma unroll
        for (int kt = 0; kt < 16; ++kt) {
            const int k0 = kt * 4 + kHalf;
            v2f a;                               // W[o][k0], W[o][k0+1]
            a.x = Ws[orow * HS + k0];
            a.y = Ws[orow * HS + k0 + 1];
            v2f b0, b1;                          // v-scaled f1 rows
            b0.x = Zs[k0 * 64 + nt0 * 16 + laneN] * vp0;
            b0.y = Zs[(k0 + 1) * 64 + nt0 * 16 + laneN] * vp0;
            b1.x = Zs[k0 * 64 + (nt0 + 1) * 16 + laneN] * vp1;
            b1.y = Zs[(k0 + 1) * 64 + (nt0 + 1) * 16 + laneN] * vp1;
            c0 = wmma4f32(a, b0, c0);
            c1 = wmma4f32(a, b1, c1);
        }
#pragma unroll
        for (int r = 0; r < 8; ++r) {            // soft-threshold + sum over p
            float cv = c0[r];
            float mag = fabsf(cv) - tp0;
            acc0[r] += (mag > 0.0f) ? copysignf(mag, cv) : 0.0f;
            cv  = c1[r];
            mag = fabsf(cv) - tp1;
            acc1[r] += (mag > 0.0f) ? copysignf(mag, cv) : 0.0f;
        }
    }

    const int rbase = ot * 16 + (lane >> 4) * 8;
    float* f6b = f6 + (size_t)b * NC * NPAD;
#pragma unroll
    for (int r = 0; r < 8; ++r) {
        f6b[(size_t)(rbase + r) * NPAD + ncol0] = acc0[r];
        f6b[(size_t)(rbase + r) * NPAD + ncol1] = acc1[r];
    }
}

// ---------------------------------------------------------------------------
// Kernel 3: inverse WHT + scale + truncate + residual add.
// One workgroup per (b,o) row: Y = H * F * H / 4096, y[:4000] = Y + x.
// ---------------------------------------------------------------------------
__global__ void __launch_bounds__(256) wht_inv(const float* __restrict__ f6,
                                               const float* __restrict__ x,
                                               float* __restrict__ y) {
    __shared__ float Xs[64 * 64];
    __shared__ float Ts[64 * HS];
    __shared__ float Hs[64 * HS];

    const int row = blockIdx.x;        // b*64 + o
    const int tid = threadIdx.x;
    const float* fr = f6 + (size_t)row * NPAD;

    fill_H(Hs, tid);
    for (int idx = tid; idx < NPAD; idx += 256)
        Xs[idx] = fr[idx];
    __syncthreads();

    const int wave  = tid >> 5;
    const int lane  = tid & 31;
    const int laneN = lane & 15;
    const int kHalf = (lane >> 4) << 1;
    const int mt    = wave & 3;
    const int nt0   = (wave >> 2) * 2;

    // ---- stage 1: T = H @ F ----
    {
        v8f c0 = {}, c1 = {};
        const int mrow = mt * 16 + laneN;
#pragma unroll
        for (int kt = 0; kt < 16; ++kt) {
            const int k0 = kt * 4 + kHalf;
            v2f a;
            a.x = Hs[mrow * HS + k0];
            a.y = Hs[mrow * HS + k0 + 1];
            v2f b0, b1;
            b0.x = Xs[k0 * 64 + nt0 * 16 + laneN];
            b0.y = Xs[(k0 + 1) * 64 + nt0 * 16 + laneN];
            b1.x = Xs[k0 * 64 + (nt0 + 1) * 16 + laneN];
            b1.y = Xs[(k0 + 1) * 64 + (nt0 + 1) * 16 + laneN];
            c0 = wmma4f32(a, b0, c0);
            c1 = wmma4f32(a, b1, c1);
        }
        const int rbase = mt * 16 + (lane >> 4) * 8;
        const int ci0   = nt0 * 16 + laneN;
#pragma unroll
        for (int r = 0; r < 8; ++r) {
            Ts[(rbase + r) * HS + ci0]      = c0[r];
            Ts[(rbase + r) * HS + ci0 + 16] = c1[r];
        }
    }
    __syncthreads();

    // ---- stage 2: Y = T @ H, scale 1/4096, truncate to 4000, + x ----
    {
        const float* xr = x + (size_t)row * NLEN;
        float* yr = y + (size_t)row * NLEN;
        const float scale = 1.0f / (float)NPAD;
        v8f c0 = {}, c1 = {};
        const int mrow = mt * 16 + laneN;
#pragma unroll
        for (int kt = 0; kt < 16; ++kt) {
            const int k0 = kt * 4 + kHalf;
            v2f a;
            a.x = Ts[mrow * HS + k0];
            a.y = Ts[mrow * HS + k0 + 1];
            v2f b0, b1;
            b0.x = Hs[k0 * HS + nt0 * 16 + laneN];
            b0.y = Hs[(k0 + 1) * HS + nt0 * 16 + laneN];
            b1.x = Hs[k0 * HS + (nt0 + 1) * 16 + laneN];
            b1.y = Hs[(k0 + 1) * HS + (nt0 + 1) * 16 + laneN];
            c0 = wmma4f32(a, b0, c0);
            c1 = wmma4f32(a, b1, c1);
        }
        const int rbase = mt * 16 + (lane >> 4) * 8;   // n1
        const int ci0   = nt0 * 16 + laneN;            // n2
#pragma unroll
        for (int r = 0; r < 8; ++r) {
            const int n0 = (rbase + r) * 64 + ci0;
            const int n1 = n0 + 16;
            if (n0 < NLEN) yr[n0] = c0[r] * scale + xr[n0];
            if (n1 < NLEN) yr[n1] = c1[r] * scale + xr[n1];
        }
    }
}

// ---------------------------------------------------------------------------
extern "C" void kernel_launch(void* const* d_in, const int* in_sizes, int n_in,
                              void* d_out, int out_size, void* d_ws, size_t ws_size,
                              hipStream_t stream) {
    const float* x = (const float*)d_in[0];   // (8,64,4000)
    const float* W = (const float*)d_in[1];   // (4,64,64)
    const float* T = (const float*)d_in[2];   // (4,4096)
    const float* v = (const float*)d_in[3];   // (4,4096)
    float* y  = (float*)d_out;                // (8,64,4000)

    float* f1 = (float*)d_ws;                       // 8*64*4096 f32 = 8.39 MB
    float* f6 = f1 + (size_t)NB * NC * NPAD;        // another 8.39 MB

    wht_fwd<<<NB * NC, 256, 0, stream>>>(x, f1);
    chmix  <<<NB * 64, 256, 0, stream>>>(f1, W, T, v, f6);
    wht_inv<<<NB * NC, 256, 0, stream>>>(f6, x, y);
}